// EdgeConditionedGraphAttention_33560874451087
// MI455X (gfx1250) — compile-verified
//
#include <hip/hip_runtime.h>
#include <hip/hip_bf16.h>

typedef __attribute__((ext_vector_type(2))) float v2f;
typedef __attribute__((ext_vector_type(8))) float v8f;
typedef __attribute__((ext_vector_type(4))) unsigned uint32x4;
typedef __attribute__((ext_vector_type(8))) int int32x8;
typedef __attribute__((ext_vector_type(4))) int int32x4;

constexpr int NN = 50000;      // nodes
constexpr int NE = 800000;     // real edges
constexpr int E2 = 850000;     // edges + self loops
constexpr int DD = 64;         // channels
constexpr int ED = 32;         // edge dim
constexpr int NH = 8;          // heads
constexpr int NT_NODE = NN / 16;   // 3125 node tiles
constexpr int NT_EDGE = E2 / 16;   // 53125 edge tiles
constexpr int WE_STRIDE = 80;      // LDS row stride (64 + 16 TDM pad), bank-conflict free

__device__ __forceinline__ v8f wmma4(v2f a, v2f b, v8f c) {
  // D = A(16x4,f32) * B(4x16,f32) + C(16x16,f32)
  return __builtin_amdgcn_wmma_f32_16x16x4_f32(
      /*neg_a=*/false, a, /*neg_b=*/false, b,
      /*c_mod=*/(short)0, c, /*reuse_a=*/false, /*reuse_b=*/false);
}

// Order-preserving float<->uint map for integer atomic max on floats.
__device__ __forceinline__ unsigned f2u_mono(float f) {
  unsigned u = __float_as_uint(f);
  return u ^ (unsigned)(((int)u >> 31) | 0x80000000);
}
__device__ __forceinline__ float u2f_mono(unsigned u) {
  return __uint_as_float(u ^ (unsigned)(((int)(~u) >> 31) | 0x80000000));
}

// ------------------- TDM: stage We[32x64] f32 into LDS with row padding ----
// D# per CDNA5 ISA ch.8: group0 {count=1, lds_addr, global_addr, type=2},
// group1 {data_size=4B, pad_enable, pad_interval=64DW, pad_amount=16DW,
//         tensor_dim0=64, tensor_dim1=32, tile_dim0=64, tile_dim1=32,
//         tensor_dim0_stride=64}. Trailing groups zero (tile_dim2=0 -> unused).
// Pad => LDS row stride 80 DW so the two lane-halves (rows r, r+2) land in
// disjoint bank sets during B-fragment ds_loads.
__device__ __forceinline__ void tdm_stage_We(const float* We, void* lds_dst) {
  unsigned lds_addr = (unsigned)(unsigned long long)lds_dst;  // low 32 bits = LDS offset
  unsigned long long ga = (unsigned long long)We;
  uint32x4 g0;
  g0[0] = 1u;                                           // count=1 (valid user D#)
  g0[1] = lds_addr;                                     // bits 63:32
  g0[2] = (unsigned)(ga & 0xFFFFFFFFull);               // global_addr[31:0]
  g0[3] = (unsigned)((ga >> 32) & 0x01FFFFFFull)        // global_addr[56:32]
        | (2u << 30);                                   // type=2 ("image")
  int32x8 g1;
  g1[0] = (int)((2u << 16)      // data_size = 4 bytes
              | (1u << 20)      // pad_enable
              | (5u << 22)      // pad_interval: 2^(5+1)=64 DWORDs
              | (15u << 25));   // pad_amount: 15+1=16 DWORDs
  g1[1] = (int)(64u << 16);     // tensor_dim0[15:0]=64 @bits 63:48
  g1[2] = (int)(32u << 16);     // tensor_dim0 hi=0 | tensor_dim1[15:0]=32 @95:80
  g1[3] = (int)(64u << 16);     // tensor_dim1 hi=0 | tile_dim0=64 @127:112
  g1[4] = 32;                   // tile_dim1=32, tile_dim2=0 (unused)
  g1[5] = 64;                   // tensor_dim0_stride[31:0]=64
  g1[6] = 0;
  g1[7] = 0;
  int32x4 z4 = {0, 0, 0, 0};
  int32x8 z8 = {0, 0, 0, 0, 0, 0, 0, 0};
  // amdgpu-toolchain (clang-23) arity: (v4u, v8i, v4i, v4i, v8i, i32 cpol)
  __builtin_amdgcn_tensor_load_to_lds(g0, g1, z4, z4, z8, 0);
  __builtin_amdgcn_s_wait_tensorcnt(0);                 // TDM done -> LDS valid
}

// ---------------------------------------------------------------- init ----
__global__ void ecga_init(float* out_sum, float* counts, float* sums,
                          unsigned* maxbits) {
  int i = blockIdx.x * blockDim.x + threadIdx.x;
  if (i < NN * DD) out_sum[i] = 0.0f;
  if (i < NN) counts[i] = 0.0f;
  if (i < NH) { sums[i] = 0.0f; maxbits[i] = 0u; }  // 0 == mapped(-inf)
}

// -------------------------------------------------------------- counts ----
__global__ void ecga_counts(const long long* __restrict__ ei, float* counts) {
  int e = blockIdx.x * blockDim.x + threadIdx.x;
  if (e < NE)        atomicAdd(&counts[(int)ei[NE + e]], 1.0f);
  else if (e < E2)   atomicAdd(&counts[e - NE], 1.0f);
}

// ------------------------------------------------------------ QKV GEMM ----
// One wave computes one 16-node tile of q,k,v. WMMA f32 16x16x4, K=64.
__global__ void ecga_qkv(const float* __restrict__ x,
                         const float* __restrict__ Wq, const float* __restrict__ bq,
                         const float* __restrict__ Wk, const float* __restrict__ bk,
                         const float* __restrict__ Wv, const float* __restrict__ bv,
                         float* q, float* k, float* v) {
  int gw   = (blockIdx.x * blockDim.x + threadIdx.x) >> 5;
  int lane = threadIdx.x & 31;
  int half = lane >> 4, lr = lane & 15;
  if (gw >= NT_NODE) return;           // wave-uniform: EXEC stays all-1
  int tile = gw;
  v2f a[16];
  const float* xr = x + (tile * 16 + lr) * DD + 2 * half;
#pragma unroll
  for (int kk = 0; kk < 16; ++kk) { a[kk].x = xr[4 * kk]; a[kk].y = xr[4 * kk + 1]; }
  const float* Ws[3] = {Wq, Wk, Wv};
  const float* bs[3] = {bq, bk, bv};
  float*       os[3] = {q, k, v};
  for (int m = 0; m < 3; ++m) {
    const float* W = Ws[m]; const float* b = bs[m]; float* o = os[m];
#pragma unroll
    for (int t = 0; t < 4; ++t) {
      v8f acc; float bb = b[t * 16 + lr];
#pragma unroll
      for (int i = 0; i < 8; ++i) acc[i] = bb;
#pragma unroll
      for (int kk = 0; kk < 16; ++kk) {
        v2f bf;
        bf.x = W[(4 * kk + 2 * half)     * DD + t * 16 + lr];
        bf.y = W[(4 * kk + 2 * half + 1) * DD + t * 16 + lr];
        acc = wmma4(a[kk], bf, acc);
      }
#pragma unroll
      for (int i = 0; i < 8; ++i)
        o[(tile * 16 + i + 8 * half) * DD + t * 16 + lr] = acc[i];
    }
  }
}

// Recompute one 16-edge x 64 tile of ee = ea@We + be into this wave's LDS slab.
// We fragments come from the TDM-staged LDS copy (stride WE_STRIDE).
__device__ __forceinline__ void ee_tile_to_lds(const float* __restrict__ ea,
                                               const float* __restrict__ WeS,
                                               const float* __restrict__ be,
                                               int e0, bool selfloop,
                                               int half, int lr, float* ee) {
  v2f a[8];
  if (!selfloop) {
    const float* ar = ea + (e0 + lr) * ED + 2 * half;
#pragma unroll
    for (int kk = 0; kk < 8; ++kk) { a[kk].x = ar[4 * kk]; a[kk].y = ar[4 * kk + 1]; }
  } else {
#pragma unroll
    for (int kk = 0; kk < 8; ++kk) { a[kk].x = 0.0f; a[kk].y = 0.0f; }
  }
#pragma unroll
  for (int t = 0; t < 4; ++t) {
    v8f acc; float bb = be[t * 16 + lr];
#pragma unroll
    for (int i = 0; i < 8; ++i) acc[i] = bb;
#pragma unroll
    for (int kk = 0; kk < 8; ++kk) {
      v2f bf;
      bf.x = WeS[(4 * kk + 2 * half)     * WE_STRIDE + t * 16 + lr];
      bf.y = WeS[(4 * kk + 2 * half + 1) * WE_STRIDE + t * 16 + lr];
      acc = wmma4(a[kk], bf, acc);
    }
#pragma unroll
    for (int i = 0; i < 8; ++i) ee[(i + 8 * half) * DD + t * 16 + lr] = acc[i];
  }
}

// --------------------------------------------- pass 1: scores + max -------
__global__ void ecga_scores(const float* __restrict__ q, const float* __restrict__ k,
                            const float* __restrict__ ea, const float* __restrict__ We,
                            const float* __restrict__ be, const long long* __restrict__ ei,
                            float* scores, unsigned* maxbits) {
  __shared__ float ee_s[8][16 * DD];
  __shared__ float WeS[ED * WE_STRIDE + 16];
  int wib = threadIdx.x >> 5, lane = threadIdx.x & 31;
  int half = lane >> 4, lr = lane & 15;
  int tile = blockIdx.x * 8 + wib;
  bool active = tile < NT_EDGE;
  int e0 = tile * 16;
  bool selfloop = (e0 >= NE);
  float* ee = &ee_s[wib][0];
  if (wib == 0) tdm_stage_We(We, &WeS[0]);        // async DMA; wave0 waits tensorcnt
  // Prefetch next block's edge_attr rows (streaming, HBM-bound).
  int pe = e0 + 128 + lane;
  if (pe < NE) __builtin_prefetch(&ea[pe * ED], 0, 1);
  __syncthreads();                                 // publish WeS to all waves
  if (active) ee_tile_to_lds(ea, &WeS[0], be, e0, selfloop, half, lr, ee);
  __syncthreads();
  if (active) {
    const float rscale = 0.35355339059327373f;  // 1/sqrt(HD)
    int h = lane & 7;
    float lmax = -3.4e38f;
#pragma unroll
    for (int p = 0; p < 4; ++p) {
      int idx = p * 32 + lane;       // (edge-in-tile, head) pair
      int el = idx >> 3;
      int eg = e0 + el;
      int s, d;
      if (selfloop) { s = eg - NE; d = s; }
      else          { s = (int)ei[eg]; d = (int)ei[NE + eg]; }
      const float4* qr = (const float4*)(q + d * DD + h * 8);
      const float4* kr = (const float4*)(k + s * DD + h * 8);
      const float*  er = ee + el * DD + h * 8;
      float4 q0 = qr[0], q1 = qr[1], k0 = kr[0], k1 = kr[1];
      float dot = q0.x * (k0.x + er[0]) + q0.y * (k0.y + er[1]) +
                  q0.z * (k0.z + er[2]) + q0.w * (k0.w + er[3]) +
                  q1.x * (k1.x + er[4]) + q1.y * (k1.y + er[5]) +
                  q1.z * (k1.z + er[6]) + q1.w * (k1.w + er[7]);
      float sc = dot * rscale;
      scores[eg * NH + h] = sc;
      lmax = fmaxf(lmax, sc);
    }
    lmax = fmaxf(lmax, __shfl_xor(lmax, 8, 32));
    lmax = fmaxf(lmax, __shfl_xor(lmax, 16, 32));
    if (lane < 8) atomicMax(&maxbits[h], f2u_mono(lmax));
  }
}

// --------------------------------------------- pass 1b: sum of exp --------
__global__ void ecga_sumexp(const float* __restrict__ scores,
                            const unsigned* __restrict__ maxbits, float* sums) {
  int lane = threadIdx.x & 31;
  int h = lane & 7;                        // stride is a multiple of 8
  float mx = u2f_mono(maxbits[h]);
  float part = 0.0f;
  int stride = blockDim.x * gridDim.x;
  for (int i = blockIdx.x * blockDim.x + threadIdx.x; i < E2 * NH; i += stride)
    part += expf(scores[i] - mx);
  part += __shfl_xor(part, 8, 32);
  part += __shfl_xor(part, 16, 32);
  if (lane < 8) atomicAdd(&sums[h], part);
}

// --------------------------------------------- pass 2: messages scatter ---
__global__ void ecga_messages(const float* __restrict__ v, const float* __restrict__ ea,
                              const float* __restrict__ We, const float* __restrict__ be,
                              const long long* __restrict__ ei,
                              const float* __restrict__ scores,
                              const unsigned* __restrict__ maxbits,
                              const float* __restrict__ sums, float* out_sum) {
  __shared__ float ee_s[8][16 * DD];
  __shared__ float al_s[8][16 * NH];
  __shared__ float WeS[ED * WE_STRIDE + 16];
  int wib = threadIdx.x >> 5, lane = threadIdx.x & 31;
  int half = lane >> 4, lr = lane & 15;
  int tile = blockIdx.x * 8 + wib;
  bool active = tile < NT_EDGE;
  int e0 = tile * 16;
  bool selfloop = (e0 >= NE);
  float* ee = &ee_s[wib][0];
  float* al = &al_s[wib][0];
  if (wib == 0) tdm_stage_We(We, &WeS[0]);
  int pe = e0 + 128 + lane;
  if (pe < NE) __builtin_prefetch(&ea[pe * ED], 0, 1);
  __syncthreads();
  if (active) {
    ee_tile_to_lds(ea, &WeS[0], be, e0, selfloop, half, lr, ee);
    int h = lane & 7;
    float mx = u2f_mono(maxbits[h]);
    float rs = 1.0f / sums[h];
#pragma unroll
    for (int p = 0; p < 4; ++p) {
      int idx = p * 32 + lane;
      int el = idx >> 3;
      al[el * NH + h] = expf(scores[(e0 + el) * NH + h] - mx) * rs;
    }
  }
  __syncthreads();
  if (active) {
#pragma unroll
    for (int el = 0; el < 16; ++el) {
      int eg = e0 + el;
      int s, d;
      if (selfloop) { s = eg - NE; d = s; }
      else          { s = (int)ei[eg]; d = (int)ei[NE + eg]; }
#pragma unroll
      for (int r = 0; r < 2; ++r) {
        int f = lane + 32 * r;
        float a = al[el * NH + (f >> 3)];
        float val = a * (v[s * DD + f] + ee[el * DD + f]);
        atomicAdd(&out_sum[d * DD + f], val);
      }
    }
  }
}

// ------------------ LN over 64 features of each row, then store -----------
__device__ __forceinline__ void layernorm_store(v8f acc[4],
                                                const float* __restrict__ g,
                                                const float* __restrict__ b,
                                                float* o, int tile, int half, int lr) {
  float mean[8], rstd[8];
#pragma unroll
  for (int i = 0; i < 8; ++i) {
    float s = acc[0][i] + acc[1][i] + acc[2][i] + acc[3][i];
#pragma unroll
    for (int dmask = 1; dmask < 16; dmask <<= 1) s += __shfl_xor(s, dmask, 32);
    mean[i] = s * (1.0f / 64.0f);
  }
#pragma unroll
  for (int i = 0; i < 8; ++i) {
    float s = 0.0f;
#pragma unroll
    for (int t = 0; t < 4; ++t) { float dl = acc[t][i] - mean[i]; s += dl * dl; }
#pragma unroll
    for (int dmask = 1; dmask < 16; dmask <<= 1) s += __shfl_xor(s, dmask, 32);
    rstd[i] = rsqrtf(s * (1.0f / 64.0f) + 1e-5f);
  }
#pragma unroll
  for (int t = 0; t < 4; ++t) {
    float gg = g[t * 16 + lr], bb = b[t * 16 + lr];
#pragma unroll
    for (int i = 0; i < 8; ++i) {
      float yn = (acc[t][i] - mean[i]) * rstd[i] * gg + bb;
      o[(tile * 16 + i + 8 * half) * DD + t * 16 + lr] = yn;
    }
  }
}

// ---------------- node epilogue: h = LN(x + (out_sum/cnt)@Wo + bo) --------
__global__ void ecga_node_out(const float* __restrict__ x, const float* __restrict__ out_sum,
                              const float* __restrict__ counts,
                              const float* __restrict__ Wo, const float* __restrict__ bo,
                              const float* __restrict__ g1, const float* __restrict__ b1,
                              float* hbuf) {
  int gw   = (blockIdx.x * blockDim.x + threadIdx.x) >> 5;
  int lane = threadIdx.x & 31;
  int half = lane >> 4, lr = lane & 15;
  if (gw >= NT_NODE) return;
  int tile = gw;
  int row = tile * 16 + lr;
  float invc = 1.0f / fmaxf(counts[row], 1.0f);
  v2f a[16];
  const float* orow = out_sum + row * DD + 2 * half;
#pragma unroll
  for (int kk = 0; kk < 16; ++kk) {
    a[kk].x = orow[4 * kk] * invc;
    a[kk].y = orow[4 * kk + 1] * invc;
  }
  v8f acc[4];
#pragma unroll
  for (int t = 0; t < 4; ++t) {
    float bb = bo[t * 16 + lr];
#pragma unroll
    for (int i = 0; i < 8; ++i)
      acc[t][i] = bb + x[(tile * 16 + i + 8 * half) * DD + t * 16 + lr];
#pragma unroll
    for (int kk = 0; kk < 16; ++kk) {
      v2f bf;
      bf.x = Wo[(4 * kk + 2 * half)     * DD + t * 16 + lr];
      bf.y = Wo[(4 * kk + 2 * half + 1) * DD + t * 16 + lr];
      acc[t] = wmma4(a[kk], bf, acc[t]);
    }
  }
  layernorm_store(acc, g1, b1, hbuf, tile, half, lr);
}

// ---------------- fused FFN: out = LN(h + gelu(h@Wf1+bf1)@Wf2 + bf2) ------
// 128 threads (4 waves); z tile kept in LDS with stride-132 padding.
__global__ void ecga_ffn(const float* __restrict__ hbuf,
                         const float* __restrict__ Wf1, const float* __restrict__ bf1,
                         const float* __restrict__ Wf2, const float* __restrict__ bf2,
                         const float* __restrict__ g2, const float* __restrict__ b2,
                         float* out) {
  __shared__ float z_s[4][16 * 132];
  int wib = threadIdx.x >> 5, lane = threadIdx.x & 31;
  int half = lane >> 4, lr = lane & 15;
  int tile = blockIdx.x * 4 + wib;
  bool active = tile < NT_NODE;
  float* zs = &z_s[wib][0];
  if (active) {
    v2f a[16];
    const float* hr = hbuf + (tile * 16 + lr) * DD + 2 * half;
#pragma unroll
    for (int kk = 0; kk < 16; ++kk) { a[kk].x = hr[4 * kk]; a[kk].y = hr[4 * kk + 1]; }
#pragma unroll
    for (int t = 0; t < 8; ++t) {         // N = 128
      v8f acc; float bb = bf1[t * 16 + lr];
#pragma unroll
      for (int i = 0; i < 8; ++i) acc[i] = bb;
#pragma unroll
      for (int kk = 0; kk < 16; ++kk) {
        v2f bf;
        bf.x = Wf1[(4 * kk + 2 * half)     * 128 + t * 16 + lr];
        bf.y = Wf1[(4 * kk + 2 * half + 1) * 128 + t * 16 + lr];
        acc = wmma4(a[kk], bf, acc);
      }
#pragma unroll
      for (int i = 0; i < 8; ++i) {
        float zv = acc[i];
        zv = 0.5f * zv * (1.0f + erff(zv * 0.70710678118654752f));  // exact gelu
        zs[(i + 8 * half) * 132 + t * 16 + lr] = zv;
      }
    }
  }
  __syncthreads();
  if (active) {
    v2f a2[32];                            // z as A-matrix, K = 128
#pragma unroll
    for (int kk = 0; kk < 32; ++kk) {
      a2[kk].x = zs[lr * 132 + 4 * kk + 2 * half];
      a2[kk].y = zs[lr * 132 + 4 * kk + 2 * half + 1];
    }
    v8f acc[4];
#pragma unroll
    for (int t = 0; t < 4; ++t) {
      float bb = bf2[t * 16 + lr];
#pragma unroll
      for (int i = 0; i < 8; ++i)
        acc[t][i] = bb + hbuf[(tile * 16 + i + 8 * half) * DD + t * 16 + lr];
#pragma unroll
      for (int kk = 0; kk < 32; ++kk) {
        v2f bf;
        bf.x = Wf2[(4 * kk + 2 * half)     * DD + t * 16 + lr];
        bf.y = Wf2[(4 * kk + 2 * half + 1) * DD + t * 16 + lr];
        acc[t] = wmma4(a2[kk], bf, acc[t]);
      }
    }
    layernorm_store(acc, g2, b2, out, tile, half, lr);
  }
}

// -------------------------------------------------------------- driver ----
extern "C" void kernel_launch(void* const* d_in, const int* in_sizes, int n_in,
                              void* d_out, int out_size, void* d_ws, size_t ws_size,
                              hipStream_t stream) {
  const float*     x   = (const float*)d_in[0];
  const long long* ei  = (const long long*)d_in[1];  // int64 [2, NE]
  const float*     ea  = (const float*)d_in[2];
  const float* Wq = (const float*)d_in[3];  const float* bq = (const float*)d_in[4];
  const float* Wk = (const float*)d_in[5];  const float* bk = (const float*)d_in[6];
  const float* Wv = (const float*)d_in[7];  const float* bv = (const float*)d_in[8];
  const float* We = (const float*)d_in[9];  const float* be = (const float*)d_in[10];
  const float* Wo = (const float*)d_in[11]; const float* bo = (const float*)d_in[12];
  const float* g1 = (const float*)d_in[13]; const float* b1 = (const float*)d_in[14];
  const float* g2 = (const float*)d_in[15]; const float* b2 = (const float*)d_in[16];
  const float* Wf1 = (const float*)d_in[17]; const float* bf1 = (const float*)d_in[18];
  const float* Wf2 = (const float*)d_in[19]; const float* bf2 = (const float*)d_in[20];
  float* out = (float*)d_out;

  float* ws      = (float*)d_ws;
  float* q       = ws;                 ws += (size_t)NN * DD;
  float* k       = ws;                 ws += (size_t)NN * DD;
  float* v       = ws;                 ws += (size_t)NN * DD;
  float* out_sum = ws;                 ws += (size_t)NN * DD;
  float* hbuf    = ws;                 ws += (size_t)NN * DD;
  float* scores  = ws;                 ws += (size_t)E2 * NH;
  float* counts  = ws;                 ws += NN;
  float* sums    = ws;                 ws += NH;
  unsigned* maxbits = (unsigned*)ws;

  ecga_init<<<dim3((NN * DD + 255) / 256), dim3(256), 0, stream>>>(out_sum, counts, sums, maxbits);
  ecga_qkv<<<dim3((NT_NODE + 7) / 8), dim3(256), 0, stream>>>(x, Wq, bq, Wk, bk, Wv, bv, q, k, v);
  ecga_counts<<<dim3((E2 + 255) / 256), dim3(256), 0, stream>>>(ei, counts);
  ecga_scores<<<dim3((NT_EDGE + 7) / 8), dim3(256), 0, stream>>>(q, k, ea, We, be, ei, scores, maxbits);
  ecga_sumexp<<<dim3(1024), dim3(256), 0, stream>>>(scores, maxbits, sums);
  ecga_messages<<<dim3((NT_EDGE + 7) / 8), dim3(256), 0, stream>>>(v, ea, We, be, ei, scores, maxbits, sums, out_sum);
  ecga_node_out<<<dim3((NT_NODE + 7) / 8), dim3(256), 0, stream>>>(x, out_sum, counts, Wo, bo, g1, b1, hbuf);
  ecga_ffn<<<dim3((NT_NODE + 3) / 4), dim3(128), 0, stream>>>(hbuf, Wf1, bf1, Wf2, bf2, g2, b2, out);
}